// FraudDetectionHybridModel_65481071407103
// MI455X (gfx1250) — compile-verified
//
#include <hip/hip_runtime.h>
#include <math.h>

// CDNA5 (gfx1250) wave32 WMMA types
typedef __attribute__((ext_vector_type(2))) float v2f;
typedef __attribute__((ext_vector_type(8))) float v8f;

// ---------------------------------------------------------------------------
// Complex helpers for the tiny unitary-builder kernel
// ---------------------------------------------------------------------------
struct Cx { float re, im; };
__device__ __forceinline__ Cx cmul(Cx a, Cx b) {
    return { a.re * b.re - a.im * b.im, a.re * b.im + a.im * b.re };
}
__device__ __forceinline__ Cx cadd(Cx a, Cx b) { return { a.re + b.re, a.im + b.im }; }

// Apply a (optionally controlled) 1-qubit gate to all 16 columns of the
// 16x16 complex matrix (ur, ui).  wire/ctrl use the reference convention:
// amplitude index bit = (3 - wire); ctrl < 0 means uncontrolled.
__device__ void apply_gate(float* ur, float* ui, int wire, int ctrl,
                           Cx m00, Cx m01, Cx m10, Cx m11)
{
    const int bit    = 3 - wire;
    const int stride = 1 << bit;
    const int cmask  = (ctrl >= 0) ? (1 << (3 - ctrl)) : 0;
    for (int col = 0; col < 16; ++col) {
        for (int i = 0; i < 16; ++i) {
            if ((i >> bit) & 1) continue;            // only pairs (i, i+stride)
            if (cmask && !(i & cmask)) continue;     // controlled: ctrl bit must be 1
            const int j = i + stride;
            Cx a = { ur[i * 16 + col], ui[i * 16 + col] };
            Cx b = { ur[j * 16 + col], ui[j * 16 + col] };
            Cx na = cadd(cmul(m00, a), cmul(m01, b));
            Cx nb = cadd(cmul(m10, a), cmul(m11, b));
            ur[i * 16 + col] = na.re;  ui[i * 16 + col] = na.im;
            ur[j * 16 + col] = nb.re;  ui[j * 16 + col] = nb.im;
        }
    }
}

// Build the full 16x16 complex circuit unitary U = G8*...*G1 from qangles.
// Writes Ur (256 floats) then Ui (256 floats) into the workspace.
__global__ void build_unitary_kernel(const float* __restrict__ q, float* __restrict__ Uout)
{
    if (blockIdx.x != 0 || threadIdx.x != 0) return;
    float ur[256], ui[256];
    for (int i = 0; i < 256; ++i) { ur[i] = 0.0f; ui[i] = 0.0f; }
    for (int i = 0; i < 16; ++i) ur[i * 16 + i] = 1.0f;

    float t, c, s;
    // OPS = rx(0), ry(1), rz(2), crx(0->1), ry(3), crx(2->3), rz(0), rx(2)
    t = q[0]; c = cosf(0.5f * t); s = sinf(0.5f * t);
    apply_gate(ur, ui, 0, -1, {c,0.f}, {0.f,-s}, {0.f,-s}, {c,0.f});        // rx w0
    t = q[1]; c = cosf(0.5f * t); s = sinf(0.5f * t);
    apply_gate(ur, ui, 1, -1, {c,0.f}, {-s,0.f}, {s,0.f}, {c,0.f});         // ry w1
    t = q[2]; c = cosf(0.5f * t); s = sinf(0.5f * t);
    apply_gate(ur, ui, 2, -1, {c,-s}, {0.f,0.f}, {0.f,0.f}, {c,s});         // rz w2
    t = q[3]; c = cosf(0.5f * t); s = sinf(0.5f * t);
    apply_gate(ur, ui, 1,  0, {c,0.f}, {0.f,-s}, {0.f,-s}, {c,0.f});        // crx c=0 t=1
    t = q[4]; c = cosf(0.5f * t); s = sinf(0.5f * t);
    apply_gate(ur, ui, 3, -1, {c,0.f}, {-s,0.f}, {s,0.f}, {c,0.f});         // ry w3
    t = q[5]; c = cosf(0.5f * t); s = sinf(0.5f * t);
    apply_gate(ur, ui, 3,  2, {c,0.f}, {0.f,-s}, {0.f,-s}, {c,0.f});        // crx c=2 t=3
    t = q[6]; c = cosf(0.5f * t); s = sinf(0.5f * t);
    apply_gate(ur, ui, 0, -1, {c,-s}, {0.f,0.f}, {0.f,0.f}, {c,s});         // rz w0
    t = q[7]; c = cosf(0.5f * t); s = sinf(0.5f * t);
    apply_gate(ur, ui, 2, -1, {c,0.f}, {0.f,-s}, {0.f,-s}, {c,0.f});        // rx w2

    for (int i = 0; i < 256; ++i) { Uout[i] = ur[i]; Uout[256 + i] = ui[i]; }
}

// ---------------------------------------------------------------------------
// Main kernel: one wave32 per 16 images.
//   Dr = Ur * S, Di = Ui * S  via 8x v_wmma_f32_16x16x4_f32 (S is real),
//   then probs = Dr^2 + Di^2, <Z0>/<Z1> via cross-half shfl, then the MLP.
// ---------------------------------------------------------------------------
__global__ __launch_bounds__(32)
void quanv_mlp_kernel(const float* __restrict__ X, const float* __restrict__ U,
                      const float* __restrict__ Wl, const float* __restrict__ bl,
                      const float* __restrict__ sc, const float* __restrict__ sf,
                      const float* __restrict__ hw, const float* __restrict__ hb,
                      float* __restrict__ out)
{
    const int lane = threadIdx.x;          // 0..31
    const int col  = lane & 15;            // matrix column / image within tile
    const int hf   = lane >> 4;            // lane half selects K (A/B) or M+8 (C)
    const int img  = blockIdx.x * 16 + col;

    // ---- A fragments for V_WMMA_F32_16X16X4_F32 (16x4 f32, 2 VGPRs) ----
    // lane L<16: A[M=L][K=4kb+0, 4kb+1];  lane L>=16: A[M=L-16][K=4kb+2, 4kb+3]
    const float* Ur = U;
    const float* Ui = U + 256;
    v2f ar[4], ai[4];
#pragma unroll
    for (int kb = 0; kb < 4; ++kb) {
        const int k0 = 4 * kb + 2 * hf;
        ar[kb][0] = Ur[col * 16 + k0];     ar[kb][1] = Ur[col * 16 + k0 + 1];
        ai[kb][0] = Ui[col * 16 + k0];     ai[kb][1] = Ui[col * 16 + k0 + 1];
    }

    // ---- initial state column for this image (purely real product state) ----
    // patch 0 pixels: x[img,0,0], x[img,0,1], x[img,1,0], x[img,1,1]
    const float* xp = X + (size_t)img * 784;
    const float p00 = xp[0], p01 = xp[1], p10 = xp[28], p11 = xp[29];
    float f[4][2];
    f[0][0] = cosf(0.5f * p00); f[0][1] = sinf(0.5f * p00);
    f[1][0] = cosf(0.5f * p01); f[1][1] = sinf(0.5f * p01);
    f[2][0] = cosf(0.5f * p10); f[2][1] = sinf(0.5f * p10);
    f[3][0] = cosf(0.5f * p11); f[3][1] = sinf(0.5f * p11);

    // ---- B fragments (4x16 f32, 2 VGPRs): lane gives S[K=4kb+2*hf+{0,1}][N=col]
    v2f sb[4];
#pragma unroll
    for (int kb = 0; kb < 4; ++kb) {
        const int a0 = 4 * kb + 2 * hf;
#pragma unroll
        for (int j = 0; j < 2; ++j) {
            const int a = a0 + j;     // amplitude index: wire0 = bit3 ... wire3 = bit0
            sb[kb][j] = f[0][(a >> 3) & 1] * f[1][(a >> 2) & 1] *
                        f[2][(a >> 1) & 1] * f[3][a & 1];
        }
    }

    // ---- complex mat-vec batch as two real 16x16x16 matmuls (K chained) ----
    v8f cr = {};   // Re(final amplitudes), 16 images wide
    v8f ci = {};   // Im(final amplitudes)
#pragma unroll
    for (int kb = 0; kb < 4; ++kb) {
        cr = __builtin_amdgcn_wmma_f32_16x16x4_f32(false, ar[kb], false, sb[kb],
                                                   (short)0, cr, false, false);
        ci = __builtin_amdgcn_wmma_f32_16x16x4_f32(false, ai[kb], false, sb[kb],
                                                   (short)0, ci, false, false);
    }

    // ---- probabilities; C/D layout: VGPR r = amp (r + 8*hf), lane&15 = image ----
    float p[8];
#pragma unroll
    for (int r = 0; r < 8; ++r) p[r] = cr[r] * cr[r] + ci[r] * ci[r];

    // wire0 sign = bit3 of amp (lane half); wire1 sign = bit2 of amp (r>=4)
    const float s0 = (p[0] + p[1] + p[2] + p[3]) + (p[4] + p[5] + p[6] + p[7]);
    const float s1 = (p[0] + p[1] + p[2] + p[3]) - (p[4] + p[5] + p[6] + p[7]);
    const float o0 = __shfl_xor(s0, 16, 32);
    const float o1 = __shfl_xor(s1, 16, 32);

    if (lane < 16) {
        float a0 = s0 - o0;   // <Z0> = sum(amps 0..7) - sum(amps 8..15)
        float a1 = s1 + o1;   // <Z1> = signed(bit2) over all amps
        // 3x: tanh(a @ W[i]^T + b[i]) * scale[i] + shift[i]
#pragma unroll
        for (int i = 0; i < 3; ++i) {
            const float n0 = tanhf(a0 * Wl[i * 4 + 0] + a1 * Wl[i * 4 + 1] + bl[i * 2 + 0])
                             * sc[i * 2 + 0] + sf[i * 2 + 0];
            const float n1 = tanhf(a0 * Wl[i * 4 + 2] + a1 * Wl[i * 4 + 3] + bl[i * 2 + 1])
                             * sc[i * 2 + 1] + sf[i * 2 + 1];
            a0 = n0; a1 = n1;
        }
        out[img] = a0 * hw[0] + a1 * hw[1] + hb[0];
    }
}

extern "C" void kernel_launch(void* const* d_in, const int* in_sizes, int n_in,
                              void* d_out, int out_size, void* d_ws, size_t ws_size,
                              hipStream_t stream)
{
    const float* x     = (const float*)d_in[0];  // (B, 28, 28)
    const float* q     = (const float*)d_in[1];  // (8,)
    const float* W     = (const float*)d_in[2];  // (3, 2, 2)
    const float* b     = (const float*)d_in[3];  // (3, 2)
    const float* scale = (const float*)d_in[4];  // (3, 2)
    const float* shift = (const float*)d_in[5];  // (3, 2)
    const float* hw    = (const float*)d_in[6];  // (1, 2)
    const float* hb    = (const float*)d_in[7];  // (1,)
    float*       out   = (float*)d_out;          // (B, 1)
    float*       U     = (float*)d_ws;           // Ur[256] + Ui[256]

    const int B = in_sizes[0] / 784;             // 4096

    build_unitary_kernel<<<1, 32, 0, stream>>>(q, U);
    quanv_mlp_kernel<<<B / 16, 32, 0, stream>>>(x, U, W, b, scale, shift, hw, hb, out);
}